// DispatchSF_24747601559670
// MI455X (gfx1250) — compile-verified
//
#include <hip/hip_runtime.h>
#include <stdint.h>

#define EXPERTS 8

typedef float v4f __attribute__((ext_vector_type(4)));
typedef int   v4i __attribute__((vector_size(16)));   // matches builtin's param type

typedef __attribute__((address_space(1))) v4i* gptr_b128;  // global (AS1)
typedef __attribute__((address_space(3))) v4i* lptr_b128;  // LDS (AS3)

// ---- CDNA5 feature detection -------------------------------------------------
#if defined(__has_builtin)
#if __has_builtin(__builtin_amdgcn_global_load_async_to_lds_b128)
#define HAVE_ASYNC_LDS 1
#endif
#if __has_builtin(__builtin_amdgcn_s_wait_asynccnt)
#define WAIT_ASYNC0() __builtin_amdgcn_s_wait_asynccnt(0)
#endif
#if __has_builtin(__builtin_amdgcn_ballot_w32)
#define HAVE_BALLOT32 1
#endif
#endif

#ifndef WAIT_ASYNC0
#define WAIT_ASYNC0() asm volatile("s_wait_asynccnt 0x0" ::: "memory")
#endif

static __device__ __forceinline__ unsigned ballot32(bool p) {
#if defined(HAVE_BALLOT32)
  return __builtin_amdgcn_ballot_w32(p);
#else
  return (unsigned)__ballot(p);   // wave32: low 32 bits
#endif
}

// ---- Phase 1: per-expert stable scan (rank = exclusive cumsum of hot_mask) ---
// One block per expert, 1024 threads = 32 wave32 waves. Builds gidx[e*N+slot]=n
// and counts[e]. Wave-level scan via v_ballot + popc (wave32-native).
__global__ __launch_bounds__(1024) void moe_scan_kernel(
    const int* __restrict__ hot_mask,   // [N, E]
    int* __restrict__ gidx,             // [E, N] slot -> token index
    int* __restrict__ counts_ws,        // [E]
    int* __restrict__ counts_out,       // [E]
    int N)
{
  const int e    = blockIdx.x;
  const int tid  = threadIdx.x;
  const int lane = tid & 31;
  const int wave = tid >> 5;

  __shared__ int wave_tot[32];
  __shared__ int wave_off[32];
  __shared__ int chunk_total;
  __shared__ int running;
  if (tid == 0) running = 0;
  __syncthreads();

  for (int base = 0; base < N; base += 1024) {
    const int n = base + tid;
    const bool pred = (n < N) && (hot_mask[(size_t)n * EXPERTS + e] != 0);

    const unsigned bal  = ballot32(pred);
    const int      rank = __popc(bal & ((1u << lane) - 1u));
    if (lane == 0) wave_tot[wave] = __popc(bal);
    __syncthreads();

    if (tid == 0) {                         // tiny serial scan over 32 waves
      int acc = 0;
      for (int w = 0; w < 32; ++w) { wave_off[w] = acc; acc += wave_tot[w]; }
      chunk_total = acc;
    }
    __syncthreads();

    if (pred) {
      const int slot = running + wave_off[wave] + rank;
      gidx[(size_t)e * N + slot] = n;
    }
    __syncthreads();
    if (tid == 0) running += chunk_total;
    __syncthreads();
  }

  if (tid == 0) {
    counts_ws[e]  = running;
    counts_out[e] = running;
  }
}

// ---- Phase 2: one block per output row (e, s) --------------------------------
// Real rows: gather x[n,:] via CDNA5 async global->LDS (ASYNCcnt), scale,
// stream out with non-temporal stores. Padding rows: stream zeros NT.
__global__ __launch_bounds__(256) void moe_gather_kernel(
    const float* __restrict__ x,        // [N, D]
    const float* __restrict__ score,    // [N, E]
    const int* __restrict__ gidx,       // [E, N]
    const int* __restrict__ counts,     // [E]
    float* __restrict__ out_data,       // [E, N, D]
    long long* __restrict__ out_tags,   // [E, N]
    int N, int D)
{
  const int row = blockIdx.x;           // row = e*N + s
  const int e   = row / N;
  const int s   = row - e * N;
  const int tid = threadIdx.x;
  const int cnt = counts[e];

  float* dst = out_data + (size_t)row * D;

  if (s >= cnt) {
    // Capacity padding: zero row, streamed past L2.
    const v4f z = {0.f, 0.f, 0.f, 0.f};
    for (int d = tid * 4; d < D; d += 256 * 4)
      __builtin_nontemporal_store(z, (v4f*)(dst + d));
    if (tid == 0) out_tags[row] = 0;
    return;
  }

  const int   n  = gidx[(size_t)e * N + s];
  const float sc = score[(size_t)n * EXPERTS + e];
  const float* src = x + (size_t)n * D;

  __shared__ float lds[1024];           // 4 KB staging (one chunk)

  for (int base = 0; base < D; base += 1024) {
    const int d = base + tid * 4;
    if (base + 1024 < D)                // prefetch next chunk (global_prefetch_b8)
      __builtin_prefetch(src + base + 1024 + tid * 4, 0, 1);

    if (d < D) {
#if defined(HAVE_ASYNC_LDS)
      // CDNA5 async VMEM: 16B global -> LDS per lane, tracked by ASYNCcnt.
      __builtin_amdgcn_global_load_async_to_lds_b128(
          (gptr_b128)(src + d),
          (lptr_b128)(&lds[tid * 4]),
          0, 0);
#else
      *(v4f*)(&lds[tid * 4]) = *(const v4f*)(src + d);
#endif
    }
#if defined(HAVE_ASYNC_LDS)
    WAIT_ASYNC0();                      // each lane reads back only its own slot
#endif
    if (d < D) {
      v4f v = *(v4f*)(&lds[tid * 4]);
      v.x *= sc; v.y *= sc; v.z *= sc; v.w *= sc;
      __builtin_nontemporal_store(v, (v4f*)(dst + d));
    }
  }

  if (tid == 0) out_tags[row] = (long long)n;
}

// ---- Host entry --------------------------------------------------------------
extern "C" void kernel_launch(void* const* d_in, const int* in_sizes, int n_in,
                              void* d_out, int out_size, void* d_ws, size_t ws_size,
                              hipStream_t stream)
{
  const float* x        = (const float*)d_in[0];
  const int*   hot_mask = (const int*)d_in[1];
  const float* score    = (const float*)d_in[2];

  const int N = in_sizes[1] / EXPERTS;      // hot_mask is [N, E]
  const int D = in_sizes[0] / N;            // x is [N, D]

  // d_out: [E,N,D] f32 data | [E,N] i64 tags | [E] i32 counts (flat, in order)
  float*     out_data   = (float*)d_out;
  long long* out_tags   = (long long*)(out_data + (size_t)EXPERTS * N * D);
  int*       out_counts = (int*)(out_tags + (size_t)EXPERTS * N);

  // Workspace: counts[E] then gidx[E*N]
  int* counts_ws = (int*)d_ws;
  int* gidx      = counts_ws + EXPERTS;

  moe_scan_kernel<<<EXPERTS, 1024, 0, stream>>>(hot_mask, gidx, counts_ws,
                                                out_counts, N);
  moe_gather_kernel<<<(unsigned)(EXPERTS * N), 256, 0, stream>>>(
      x, score, gidx, counts_ws, out_data, out_tags, N, D);
}